// OT_68341519614838
// MI455X (gfx1250) — compile-verified
//
#include <hip/hip_runtime.h>

typedef __attribute__((ext_vector_type(2))) float v2f;
typedef __attribute__((ext_vector_type(8))) float v8f;

#define FEA      64
#define BS       16
#define NPATCH   42
#define NVARS    21
#define NMAT     (BS * NVARS * NPATCH)      // 14112
#define TX_SIZE  (BS * NPATCH * NVARS * FEA) // 903168
#define LN_EPS   1e-5f

// One block (4 wave32) per matrix n. Wave w owns rows [16w, 16w+16).
// Fused: GEMM (WMMA f32 16x16x4) -> +bias -> LayerNorm -> softmax -> xf@Ps -> cost partial.
__global__ __launch_bounds__(128)
void ot_fused_kernel(const float* __restrict__ x,
                     const float* __restrict__ P,
                     const float* __restrict__ mask,
                     const float* __restrict__ W,
                     const float* __restrict__ bias,
                     const float* __restrict__ gamma,
                     const float* __restrict__ beta,
                     float* __restrict__ out,       // tx region of d_out
                     float* __restrict__ cost_ws)   // per-matrix cost partial
{
    // W^T staged in WMMA B-matrix layout: [ks][nt][lane][2]
    __shared__ float sB[16 * 4 * 32 * 2];
    __shared__ float sXf[64];
    __shared__ float sTx[4][64];
    __shared__ float sCost[4];

    const int tid  = threadIdx.x;
    const int n    = blockIdx.x;
    const int w    = tid >> 5;        // wave id 0..3
    const int lane = tid & 31;
    const int lo   = lane & 15;       // lane within half
    const int hh   = lane >> 4;       // half id

    // ---- Stage B = W^T in WMMA B layout: B[k][nc] = W[nc][k]
    // sB element e: v=e&1, lane=(e>>1)&31, nt=(e>>6)&3, ks=e>>8
    // k = ks*4 + v + 2*(lane>=16), nc = nt*16 + lane%16
    for (int e = tid; e < 16 * 4 * 32 * 2; e += 128) {
        int v  = e & 1;
        int ln = (e >> 1) & 31;
        int nt = (e >> 6) & 3;
        int ks = e >> 8;
        int l2 = ln & 15, h2 = ln >> 4;
        int k  = ks * 4 + v + 2 * h2;
        int nc = nt * 16 + l2;
        sB[e] = W[nc * 64 + k];
    }

    // ---- decode n -> (b, var, patch); xf[n][f] = x[b][p][vv][f]; out uses same address
    const int bidx = n / (NVARS * NPATCH);
    const int rem  = n % (NVARS * NPATCH);
    const int vv   = rem / NPATCH;
    const int pp   = rem % NPATCH;
    const long xbase = ((long)(bidx * NPATCH + pp) * NVARS + vv) * FEA;
    if (tid < 64) sXf[tid] = x[xbase + tid];
    __syncthreads();

    // ---- per-lane column parameters (col = nt*16 + lo)
    float bv[4], gv[4], btv[4], mv[4];
    #pragma unroll
    for (int nt = 0; nt < 4; ++nt) {
        int c = nt * 16 + lo;
        bv[nt]  = bias[c];
        gv[nt]  = gamma[c];
        btv[nt] = beta[c];
        mv[nt]  = mask[(long)n * FEA + c];
    }

    // ---- GEMM: Pl rows [16w,16w+16) = P_rows @ W^T via V_WMMA_F32_16X16X4_F32
    v8f acc[4] = {};
    const float* Pn  = P + (long)n * FEA * FEA;
    const int    row = 16 * w + lo;            // A: lanes 0-15 hold M=0..15
    #pragma unroll
    for (int ks = 0; ks < 16; ++ks) {
        // A tile: VGPR0/1 = K pair; low half K=0,1  high half K=2,3 (of this 4-step)
        v2f a = *(const v2f*)(Pn + row * 64 + ks * 4 + 2 * hh);
        #pragma unroll
        for (int nt = 0; nt < 4; ++nt) {
            v2f bb = *(const v2f*)&sB[((ks * 4 + nt) * 32 + lane) * 2];
            acc[nt] = __builtin_amdgcn_wmma_f32_16x16x4_f32(
                false, a, false, bb, (short)0, acc[nt], false, false);
        }
    }

    // ---- fused bias + LayerNorm + softmax, per row (VGPR element v = row v / v+8)
    #pragma unroll
    for (int v = 0; v < 8; ++v) {
        float xel[4];
        #pragma unroll
        for (int nt = 0; nt < 4; ++nt) xel[nt] = acc[nt][v] + bv[nt];

        float s1 = xel[0] + xel[1] + xel[2] + xel[3];
        float s2 = xel[0]*xel[0] + xel[1]*xel[1] + xel[2]*xel[2] + xel[3]*xel[3];
        #pragma unroll
        for (int m = 1; m < 16; m <<= 1) {
            s1 += __shfl_xor(s1, m, 32);
            s2 += __shfl_xor(s2, m, 32);
        }
        float mu  = s1 * (1.0f / 64.0f);
        float var = s2 * (1.0f / 64.0f) - mu * mu;
        float rs  = rsqrtf(var + LN_EPS);

        float y[4];
        float mx = -3.402823466e38f;
        #pragma unroll
        for (int nt = 0; nt < 4; ++nt) {
            y[nt] = (xel[nt] - mu) * rs * gv[nt] + btv[nt];
            mx = fmaxf(mx, y[nt]);
        }
        #pragma unroll
        for (int m = 1; m < 16; m <<= 1) mx = fmaxf(mx, __shfl_xor(mx, m, 32));

        float es = 0.0f;
        #pragma unroll
        for (int nt = 0; nt < 4; ++nt) { y[nt] = __expf(y[nt] - mx); es += y[nt]; }
        #pragma unroll
        for (int m = 1; m < 16; m <<= 1) es += __shfl_xor(es, m, 32);

        float inv = 1.0f / es;
        #pragma unroll
        for (int nt = 0; nt < 4; ++nt) acc[nt][v] = y[nt] * inv;  // Ps in registers
    }

    // ---- tx partial (per col) and cost partial
    float txp[4] = {0.f, 0.f, 0.f, 0.f};
    float costp = 0.0f;
    #pragma unroll
    for (int nt = 0; nt < 4; ++nt) {
        float ps_sum = 0.0f;
        #pragma unroll
        for (int v = 0; v < 8; ++v) {
            float pval = acc[nt][v];                  // row 16w + 8*hh + v, col nt*16+lo
            txp[nt] += sXf[16 * w + 8 * hh + v] * pval;
            ps_sum  += pval;
        }
        costp += mv[nt] * ps_sum;
    }
    // combine the two halves (same col, different rows)
    #pragma unroll
    for (int nt = 0; nt < 4; ++nt) txp[nt] += __shfl_xor(txp[nt], 16, 32);
    // full-wave cost reduction
    #pragma unroll
    for (int m = 1; m < 32; m <<= 1) costp += __shfl_xor(costp, m, 32);

    if (hh == 0) {
        #pragma unroll
        for (int nt = 0; nt < 4; ++nt) sTx[w][nt * 16 + lo] = txp[nt];
    }
    if (lane == 0) sCost[w] = costp;
    __syncthreads();

    if (tid < 64)
        out[xbase + tid] = sTx[0][tid] + sTx[1][tid] + sTx[2][tid] + sTx[3][tid];
    if (tid == 0)
        cost_ws[n] = sCost[0] + sCost[1] + sCost[2] + sCost[3];
}

// Deterministic fixed-order reduction of per-matrix cost partials.
__global__ __launch_bounds__(256)
void cost_reduce_kernel(const float* __restrict__ ws, float* __restrict__ cost_out,
                        int n, float scale)
{
    __shared__ float s[256];
    float acc = 0.0f;
    for (int i = threadIdx.x; i < n; i += 256) acc += ws[i];
    s[threadIdx.x] = acc;
    __syncthreads();
    for (int st = 128; st > 0; st >>= 1) {
        if (threadIdx.x < st) s[threadIdx.x] += s[threadIdx.x + st];
        __syncthreads();
    }
    if (threadIdx.x == 0) cost_out[0] = s[0] * scale;
}

extern "C" void kernel_launch(void* const* d_in, const int* in_sizes, int n_in,
                              void* d_out, int out_size, void* d_ws, size_t ws_size,
                              hipStream_t stream)
{
    const float* x     = (const float*)d_in[0];
    const float* P     = (const float*)d_in[1];
    const float* mask  = (const float*)d_in[2];
    const float* W     = (const float*)d_in[3];
    const float* b     = (const float*)d_in[4];
    const float* gamma = (const float*)d_in[5];
    const float* beta  = (const float*)d_in[6];
    float* out = (float*)d_out;
    float* ws  = (float*)d_ws;   // NMAT floats of cost partials

    ot_fused_kernel<<<NMAT, 128, 0, stream>>>(x, P, mask, W, b, gamma, beta, out, ws);
    cost_reduce_kernel<<<1, 256, 0, stream>>>(ws, out + TX_SIZE, NMAT,
                                              1.0f / ((float)NMAT * 64.0f * 64.0f));
}